// ExtendedKalmanFilter_83726092468660
// MI455X (gfx1250) — compile-verified
//
#include <hip/hip_runtime.h>

// EKF scan for MI455X (gfx1250, wave32).
//  - 256-thread block stages measurements/true_positions (64KB) into LDS.
//  - Wave 0 keeps the 5x5 covariance P resident in the WMMA 16x16 f32
//    accumulator layout; the rank-2 Riccati update P <- P - K*P[0:2,:]
//    is ONE v_wmma_f32_16x16x4_f32 per step.
//  - 2x2 S-inverse runs on the scalar float pipe via readlane broadcasts;
//    v_rcp_f32 (raw, ~1ulp) instead of IEEE divide to shorten the
//    loop-carried dependency chain.
//  - Negation folded into the inverse coefficients (scalar side) so the
//    WMMA A-operand (-K^T) comes directly from two vector FMAs.

typedef float v2f __attribute__((ext_vector_type(2)));
typedef float v8f __attribute__((ext_vector_type(8)));

#define EKF_STEPS 4096

__device__ __forceinline__ float rdlane(float v, int lane) {
  int i = __builtin_amdgcn_readlane(__builtin_bit_cast(int, v), lane);
  return __builtin_bit_cast(float, i);
}

__global__ __launch_bounds__(256) void ekf_scan_kernel(
    const float* __restrict__ meas,   // (STEPS,2)
    const float* __restrict__ tpos,   // (STEPS,2)
    const float* __restrict__ X0,     // (5,1)
    const float* __restrict__ Qm,     // (1,)
    const float* __restrict__ R,      // (2,2)
    float* __restrict__ out,          // (1,)
    int steps)
{
  __shared__ float lm[2 * EKF_STEPS];
  __shared__ float lt[2 * EKF_STEPS];

  const int tid = threadIdx.x;
  if (steps > EKF_STEPS) steps = EKF_STEPS;
  const int n4 = (2 * steps) / 4;   // float4 count per array

  // Cooperative LDS staging with 128-bit loads (all 8 waves).
  const float4* m4 = (const float4*)meas;
  const float4* t4 = (const float4*)tpos;
  float4* lm4 = (float4*)lm;
  float4* lt4 = (float4*)lt;
  for (int i = tid; i < n4; i += 256) {
    lm4[i] = m4[i];
    lt4[i] = t4[i];
  }
  __syncthreads();

  if (tid >= 32) return;            // waves 1..7 retire; wave 0 has EXEC all-1s
  const int lane = tid;

  // ---- P0 = 100*I in WMMA C/D layout: P(m,n) at lane n, vgpr m ----
  v8f c = {};
  c[0] = (lane == 0) ? 100.0f : 0.0f;
  c[1] = (lane == 1) ? 100.0f : 0.0f;
  c[2] = (lane == 2) ? 100.0f : 0.0f;
  c[3] = (lane == 3) ? 100.0f : 0.0f;
  c[4] = (lane == 4) ? 100.0f : 0.0f;

  // ---- X striped across lanes: lane m holds X(m), lanes >=5 zero ----
  float xv = 0.0f;
  if (lane < 5) xv = X0[lane];

  const float q   = Qm[0];
  const float r00 = R[0], r01 = R[1], r10 = R[2], r11 = R[3];

  // Loop-invariant diagonal-add masks for P += q*I.
  const float qm0 = (lane == 0) ? q : 0.0f;
  const float qm1 = (lane == 1) ? q : 0.0f;
  const float qm2 = (lane == 2) ? q : 0.0f;
  const float qm3 = (lane == 3) ? q : 0.0f;
  const float qm4 = (lane == 4) ? q : 0.0f;

  // Uniform broadcasts of X(0), X(1) (predict never touches rows 0,1).
  float u0 = rdlane(xv, 0);
  float u1 = rdlane(xv, 1);

  float loss = 0.0f;

  for (int k = 0; k < steps; ++k) {
    const float2 z  = *(const float2*)&lm[2 * k];
    const float2 tp = *(const float2*)&lt[2 * k];

    // --- predict X: X(3) += X(4) ---
    const float x4u = rdlane(xv, 4);
    xv += (lane == 3) ? x4u : 0.0f;

    // --- predict P: P += q*I (diagonal lanes only) ---
    c[0] += qm0; c[1] += qm1; c[2] += qm2; c[3] += qm3; c[4] += qm4;

    // --- S = P[0:2,0:2] + R ; 2x2 inverse (scalar float pipe) ---
    const float s00 = rdlane(c[0], 0) + r00;
    const float s01 = rdlane(c[0], 1) + r01;
    const float s10 = rdlane(c[1], 0) + r10;
    const float s11 = rdlane(c[1], 1) + r11;
    const float det  = s00 * s11 - s01 * s10;
    // Raw hardware reciprocal (~1 ulp): det ~ O(1e4), ample precision, and
    // removes the 7-op IEEE divide sequence from the serial chain.
    const float rdet  = __builtin_amdgcn_rcpf(det);
    const float nrdet = -rdet;
    // NEGATED inverse coefficients: n* = -invS row-combos, so the gain FMAs
    // directly produce the WMMA A-operand (-K^T).
    const float ni00 = s11 * nrdet;   // = -i00
    const float ni10 = s10 * rdet;    // = -i10
    const float ni01 = s01 * rdet;    // = -i01
    const float ni11 = s00 * nrdet;   // = -i11

    // --- -K^T rows, striped over lanes (uses P symmetry: P(m,0)=c0(m)) ---
    // n0(m) = -K(m,0), n1(m) = -K(m,1); zero for lanes >= 5 and 16..31.
    const float n0 = fmaf(c[0], ni00, c[1] * ni10);
    const float n1 = fmaf(c[0], ni01, c[1] * ni11);

    // --- X update: X += K*Y  ==  X += (-K)*(X[:2]-z) ---
    const float ny0 = u0 - z.x;
    const float ny1 = u1 - z.y;
    xv = fmaf(n0, ny0, fmaf(n1, ny1, xv));

    // --- Riccati update in ONE full-precision WMMA:
    //     D = (-K^T) x (P rows 0,1) + P  ==  P - K * P[0:2,:]
    // A (16x4 f32): lane m holds A(m,0..1); K-cols 2,3 (lanes 16-31) are
    // zero because tile rows 8,9 of P are zero.
    v2f a; a[0] = n0;   a[1] = n1;
    v2f b; b[0] = c[0]; b[1] = c[1];
    c = __builtin_amdgcn_wmma_f32_16x16x4_f32(
        /*neg_a=*/false, a, /*neg_b=*/false, b,
        /*c_mod=*/(short)0, c, /*reuse_a=*/false, /*reuse_b=*/false);

    // --- prediction = post-update X[0:2] (f() only modifies row 3) ---
    u0 = rdlane(xv, 0);
    u1 = rdlane(xv, 1);
    const float e0 = u0 - tp.x;
    const float e1 = u1 - tp.y;
    loss = fmaf(e0, e0, fmaf(e1, e1, loss));
  }

  if (lane == 0) out[0] = loss / (float)(2 * steps);
}

extern "C" void kernel_launch(void* const* d_in, const int* in_sizes, int n_in,
                              void* d_out, int out_size, void* d_ws, size_t ws_size,
                              hipStream_t stream) {
  (void)n_in; (void)out_size; (void)d_ws; (void)ws_size;
  const float* meas = (const float*)d_in[0];
  const float* tpos = (const float*)d_in[1];
  const float* X0   = (const float*)d_in[2];
  const float* Qm   = (const float*)d_in[3];
  const float* R    = (const float*)d_in[4];
  float* out = (float*)d_out;
  const int steps = in_sizes[0] / 2;

  ekf_scan_kernel<<<1, 256, 0, stream>>>(meas, tpos, X0, Qm, R, out, steps);
}